// AdaptiveGraphPropagation_26207890440715
// MI455X (gfx1250) — compile-verified
//
#include <hip/hip_runtime.h>
#include <cstddef>
#include <cstdint>

#define KPARTS 6
#define BATCH  256
#define DIM    768
#define NMEM   32768

typedef __attribute__((ext_vector_type(2))) float v2f;
typedef __attribute__((ext_vector_type(8))) float v8f;

// ---------------------------------------------------------------------------
// 1) reciprocal L2 norm per row: one 256-thread block per row of length 768
// ---------------------------------------------------------------------------
__global__ void rownorm_kernel(const float* __restrict__ x, float* __restrict__ rn)
{
    const int row = blockIdx.x;
    const int t   = threadIdx.x;
    const float* p = x + (size_t)row * DIM;
    float a0 = p[t], a1 = p[t + 256], a2 = p[t + 512];
    float ss = a0 * a0 + a1 * a1 + a2 * a2;
    __shared__ float sb[256];
    sb[t] = ss; __syncthreads();
    for (int s = 128; s > 0; s >>= 1) { if (t < s) sb[t] += sb[t + s]; __syncthreads(); }
    if (t == 0) rn[row] = 1.0f / fmaxf(sqrtf(sb[0]), 1e-12f);
}

// ---------------------------------------------------------------------------
// 2) sim GEMM: V_WMMA_F32_16X16X4_F32 + double-buffered async global->LDS
//    block: 256 threads = 8 waves (2x4), tile 32(M) x 64(N), K-chunks of 256
// ---------------------------------------------------------------------------
#define BM 32
#define BN 64
#define KC 256
#define LDA (KC + 4)   // 260 floats: %64==4 -> bank-conflict-free, 1040B = 16B multiple

// async copy of a ROWS x KC f32 slab (global, row stride DIM) into LDS.
// GVS mode: uniform 64-bit SGPR base + per-lane 32-bit byte offset.
// vdst carries the LDS byte offset (low 32 bits of the generic pointer).
template <int ROWS>
__device__ __forceinline__ void stage_async(float (*dst)[LDA],
                                            const float* __restrict__ g,
                                            int kc, int tid)
{
    #pragma unroll
    for (int i = 0; i < (ROWS * KC / 4) / 256; ++i) {
        int idx = tid + i * 256;
        int r = idx >> 6;           // KC/4 == 64 float4 per row
        int c = (idx & 63) << 2;    // float column
        unsigned loff = (unsigned)(uintptr_t)(&dst[r][c]);
        unsigned goff = (unsigned)((r * DIM + kc + c) * sizeof(float));
        asm volatile("global_load_async_to_lds_b128 %0, %1, %2"
                     :: "v"(loff), "v"(goff), "s"(g)
                     : "memory");
    }
}

__device__ __forceinline__ void wait_async_all()
{
    asm volatile("s_wait_asynccnt 0x0" ::: "memory");
}

__global__ __launch_bounds__(256) void simgemm_kernel(
    const float* __restrict__ f,    // [K,B,D]
    const float* __restrict__ mem,  // [K,N,D]
    const float* __restrict__ rnf,  // [K,B]
    const float* __restrict__ rnm,  // [K,N]
    float* __restrict__ sim)        // [K,B,N]
{
    __shared__ float As[2][BM][LDA];   //  66,560 B
    __shared__ float Bs[2][BN][LDA];   // 133,120 B  (total 199,680 of 320K/WGP)

    const int kp    = blockIdx.z;
    const int mbase = blockIdx.y * BM;
    const int nbase = blockIdx.x * BN;
    const int tid   = threadIdx.x;
    const int wave  = tid >> 5;
    const int lane  = tid & 31;
    const int half  = lane >> 4;    // 0: K pair {0,1}, 1: K pair {2,3}
    const int l     = lane & 15;    // M (for A) / N (for B) within fragment
    const int wm    = wave >> 2;    // 0..1
    const int wn    = wave & 3;     // 0..3

    const float* Ag = f   + (size_t)(kp * BATCH + mbase) * DIM;
    const float* Bg = mem + (size_t)(kp * NMEM  + nbase) * DIM;

    v8f acc = {};

    const int NCHUNK = DIM / KC;    // 3

    // prologue: fill buffer 0
    stage_async<BM>(As[0], Ag, 0, tid);
    stage_async<BN>(Bs[0], Bg, 0, tid);
    wait_async_all();
    __syncthreads();

    int cur = 0;
    for (int ic = 0; ic < NCHUNK; ++ic) {
        // prefetch next K-chunk into the other buffer while we compute
        if (ic + 1 < NCHUNK) {
            stage_async<BM>(As[cur ^ 1], Ag, (ic + 1) * KC, tid);
            stage_async<BN>(Bs[cur ^ 1], Bg, (ic + 1) * KC, tid);
        }

        // A fragment (16x4, M=l, K=2*half+v), B fragment (4x16, N=l, K=2*half+v)
        const float* arow = &As[cur][wm * 16 + l][2 * half];
        const float* brow = &Bs[cur][wn * 16 + l][2 * half];
        #pragma unroll 8
        for (int kk = 0; kk < KC; kk += 4) {
            v2f a = *(const v2f*)(arow + kk);
            v2f b = *(const v2f*)(brow + kk);
            acc = __builtin_amdgcn_wmma_f32_16x16x4_f32(
                false, a, false, b, (short)0, acc, false, false);
        }

        if (ic + 1 < NCHUNK) {
            wait_async_all();       // my async fills of buf^1 have landed
            __syncthreads();        // everyone's fills landed / reads of cur done
            cur ^= 1;
        }
    }

    // epilogue: D element v maps to (M = v + 8*half, N = l) within the tile
    const int   ncol = nbase + wn * 16 + l;
    const float sn   = rnm[kp * NMEM + ncol];
    const int   row0 = mbase + wm * 16 + half * 8;
    #pragma unroll
    for (int v = 0; v < 8; ++v) {
        const int r = row0 + v;
        sim[(size_t)(kp * BATCH + r) * NMEM + ncol] = acc[v] * rnf[kp * BATCH + r] * sn;
    }
}

// ---------------------------------------------------------------------------
// 3) top-5 + softmax scatter + entropy weight (exp(-H)); one block per row
// ---------------------------------------------------------------------------
__global__ __launch_bounds__(256) void topk_kernel(
    const float* __restrict__ sim,  // [K*B, N]
    float* __restrict__ soft,       // [K*B, N], pre-zeroed
    float* __restrict__ wexp)       // [K*B]
{
    const int row  = blockIdx.x;
    const int t    = threadIdx.x;
    const int lane = t & 31;
    const int wave = t >> 5;
    const float* s = sim + (size_t)row * NMEM;

    float v[5]; int id[5];
    #pragma unroll
    for (int j = 0; j < 5; ++j) { v[j] = -3.0e38f; id[j] = -1; }

    for (int n = t; n < NMEM; n += 256) {
        float x = s[n];
        if (x > v[4]) {
            v[4] = x; id[4] = n;
            #pragma unroll
            for (int j = 4; j > 0; --j) {
                if (v[j] > v[j - 1]) {
                    float tv = v[j]; v[j] = v[j - 1]; v[j - 1] = tv;
                    int   ti = id[j]; id[j] = id[j - 1]; id[j - 1] = ti;
                }
            }
        }
    }

    // wave-level top-5: 5 rounds of argmax-pop over sorted per-lane lists
    __shared__ float sv[8 * 5];
    __shared__ int   si[8 * 5];
    int h = 0;
    #pragma unroll
    for (int r = 0; r < 5; ++r) {
        float best = (h < 5) ? v[h] : -3.0e38f;
        int   bl   = lane;
        #pragma unroll
        for (int off = 16; off > 0; off >>= 1) {
            float ob  = __shfl_xor(best, off, 32);
            int   obl = __shfl_xor(bl,   off, 32);
            if (ob > best || (ob == best && obl < bl)) { best = ob; bl = obl; }
        }
        int myi  = (h < 5) ? id[h] : -1;
        int widx = __shfl(myi, bl, 32);
        if (lane == 0) { sv[wave * 5 + r] = best; si[wave * 5 + r] = widx; }
        if (lane == bl) h++;
    }
    __syncthreads();

    if (t == 0) {
        float fv[5]; int fi[5];
        #pragma unroll
        for (int j = 0; j < 5; ++j) { fv[j] = -3.0e38f; fi[j] = -1; }
        for (int c = 0; c < 40; ++c) {
            float x = sv[c]; int ix = si[c];
            if (x > fv[4]) {
                fv[4] = x; fi[4] = ix;
                #pragma unroll
                for (int j = 4; j > 0; --j) {
                    if (fv[j] > fv[j - 1]) {
                        float tv = fv[j]; fv[j] = fv[j - 1]; fv[j - 1] = tv;
                        int   ti = fi[j]; fi[j] = fi[j - 1]; fi[j - 1] = ti;
                    }
                }
            }
        }
        // softmax over the 5 survivors; all other entries underflow to 0.0f
        const float vmax = fv[0];
        float e[5], z = 0.f;
        #pragma unroll
        for (int j = 0; j < 5; ++j) { e[j] = expf((fv[j] - vmax) * (1.0f / 3.0f)); z += e[j]; }
        float ent = 0.f;
        #pragma unroll
        for (int j = 0; j < 5; ++j) {
            float p = e[j] / z;
            soft[(size_t)row * NMEM + fi[j]] = p;
            float pc = fmaxf(p, 1e-8f);
            ent -= pc * logf(pc);
        }
        // clipped zeros: (N-5) * (-1e-8 * ln(1e-8))
        ent += (float)(NMEM - 5) * 1.8420680744e-7f;
        wexp[row] = expf(-ent);
    }
}

// ---------------------------------------------------------------------------
// 4) w = exp(-H) / (mean_b exp(-H) + 1e-8): one block per part k
// ---------------------------------------------------------------------------
__global__ void wnorm_kernel(const float* __restrict__ wexp, float* __restrict__ wout)
{
    const int kp = blockIdx.x;
    const int t  = threadIdx.x;
    float e = wexp[kp * BATCH + t];
    __shared__ float sb[256];
    sb[t] = e; __syncthreads();
    for (int s = 128; s > 0; s >>= 1) { if (t < s) sb[t] += sb[t + s]; __syncthreads(); }
    float mean = sb[0] * (1.0f / BATCH);
    wout[kp * BATCH + t] = e / (mean + 1e-8f);
}

// ---------------------------------------------------------------------------
extern "C" void kernel_launch(void* const* d_in, const int* in_sizes, int n_in,
                              void* d_out, int out_size, void* d_ws, size_t ws_size,
                              hipStream_t stream)
{
    (void)in_sizes; (void)n_in; (void)out_size; (void)ws_size;
    const float* pf  = (const float*)d_in[0];   // [K,B,D]
    const float* mem = (const float*)d_in[1];   // [K,N,D]

    float* out  = (float*)d_out;
    const size_t KBN = (size_t)KPARTS * BATCH * NMEM;
    float* soft = out;            // [K,B,N]
    float* sim  = out + KBN;      // [K,B,N]
    float* wout = out + 2 * KBN;  // [K,B]

    float* rnm  = (float*)d_ws;                          // K*N floats
    float* rnf  = rnm + (size_t)KPARTS * NMEM;           // K*B floats
    float* wexp = rnf + (size_t)KPARTS * BATCH;          // K*B floats

    hipMemsetAsync(soft, 0, KBN * sizeof(float), stream);

    rownorm_kernel<<<KPARTS * BATCH, 256, 0, stream>>>(pf, rnf);
    rownorm_kernel<<<KPARTS * NMEM, 256, 0, stream>>>(mem, rnm);

    dim3 grid(NMEM / BN, BATCH / BM, KPARTS);
    simgemm_kernel<<<grid, 256, 0, stream>>>(pf, mem, rnf, rnm, sim);

    topk_kernel<<<KPARTS * BATCH, 256, 0, stream>>>(sim, soft, wexp);
    wnorm_kernel<<<KPARTS, 256, 0, stream>>>(wexp, wout);
}